// MXQuantizer_76312978915926
// MI455X (gfx1250) — compile-verified
//
#include <hip/hip_runtime.h>

// MX-FP4 (e2m1) fake-quantization of 8192x8192 f32, CDNA5 (gfx1250) wave32.
// Pipeline: GLOBAL_LOAD_ASYNC_TO_LDS_B128 double-buffered row staging,
// s_wait_asynccnt synchronization, XOR-swizzled LDS layout for conflict-free
// ds_load_b128, thread-per-32-block quantization, b128 global stores.

#define ROW_LEN   8192          // floats per row
#define ROW_BYTES (ROW_LEN * 4) // 32 KB
#define THREADS   256           // 8 wave32s
#define NUM_WG    1024

__device__ __forceinline__ unsigned f2u(float x) {
    return __builtin_bit_cast(unsigned, x);
}
// floor(log2(x)) for positive normal x (exponent field extract)
__device__ __forceinline__ int ilogb_pos(float x) {
    return (int)((f2u(x) >> 23) & 0xFF) - 127;
}

// Swizzle of 16B-chunk index within a 32KB row buffer (2048 chunks).
// Permutes chunks inside each 256B (16-chunk) group so that both the async
// DMA writes (lane-contiguous) and the ds_load_b128 reads (lane stride 128B)
// hit all 64 LDS banks conflict-free.
__device__ __forceinline__ unsigned swz(unsigned c) {
    return (c & ~15u) | ((c ^ (c >> 3)) & 15u);
}

// Async copy 16B global -> LDS (per lane). lds_off is a byte offset from the
// workgroup LDS base (low 32 bits of the flat pointer to __shared__ memory).
__device__ __forceinline__ void async_load16(unsigned lds_off, const void* g) {
    asm volatile("global_load_async_to_lds_b128 %0, %1, off"
                 :: "v"(lds_off), "v"(g)
                 : "memory");
}
#define S_WAIT_ASYNC_8() asm volatile("s_wait_asynccnt 8" ::: "memory")
#define S_WAIT_ASYNC_0() asm volatile("s_wait_asynccnt 0" ::: "memory")

// e2m1 element quantize-dequantize. inv_s = inv_nano * 2^-shared_exp,
// outmul = m_nano2 * 2^shared_exp (both exact powers of two).
__device__ __forceinline__ float qelem(float x, float inv_s, float outmul) {
    float A = x * inv_s;
    float a = fabsf(A);
    int pe = 0;
    if (a > 0.f) {
        pe = ilogb_pos(a);
        if (pe < 0) pe = 0;               // min private exp = 0 for e2m1
    }
    float r   = floorf(ldexpf(a, 1 - pe) + 0.5f); // round mantissa (1 expl. bit)
    float res = r * ldexpf(0.5f, pe);
    res = fminf(res, 6.f);                // saturate to max normal
    return copysignf(res, A) * outmul;
}

__global__ void __launch_bounds__(THREADS)
mxq_kernel(const float* __restrict__ x, float* __restrict__ out, int nrows) {
    __shared__ __align__(16) float smem[2 * ROW_LEN + 16];
    // Low 32 bits of the flat pointer to LDS == byte offset from LDS base.
    const unsigned lds_base = (unsigned)(unsigned long long)(void*)smem;

    const int t = threadIdx.x;
    const int w = t >> 5;    // wave id 0..7
    const int l = t & 31;    // lane id
    const int stride = gridDim.x;

    // Wave w stages chunks [256w, 256w+256) of a row: exactly the data its
    // own 32 threads later read, so no cross-wave barrier is needed for data.
    auto issue_row = [&](int row, int buf) {
        const float* g = x + (size_t)row * ROW_LEN;
        const unsigned base = (unsigned)(w * 256);
#pragma unroll
        for (int k = 0; k < 8; ++k) {
            unsigned c  = base + (unsigned)(k * 32 + l); // chunk index
            unsigned pc = swz(c);
            async_load16(lds_base + (unsigned)buf * ROW_BYTES + pc * 16u,
                         g + (size_t)c * 4u);
        }
    };

    int r = blockIdx.x;
    if (r >= nrows) return;
    issue_row(r, 0);

    int i = 0;
    for (; r < nrows; r += stride, ++i) {
        const int p  = i & 1;
        const int rn = r + stride;
        if (rn < nrows) { issue_row(rn, 1 - p); S_WAIT_ASYNC_8(); }
        else            { S_WAIT_ASYNC_0(); }

        // --- load this thread's 32-element MX block from swizzled LDS ---
        float4 v[8];
        float bmax = 0.f;
#pragma unroll
        for (int k = 0; k < 8; ++k) {
            unsigned c  = (unsigned)(t * 8 + k);
            unsigned pc = swz(c);
            float4 q = *(const float4*)(smem + p * ROW_LEN + pc * 4u);
            v[k] = q;
            bmax = fmaxf(bmax, fmaxf(fmaxf(fabsf(q.x), fabsf(q.y)),
                                     fmaxf(fabsf(q.z), fabsf(q.w))));
        }

        // --- row max: shfl-xor wave reduce + 8-entry LDS combine ---
        float wmax = bmax;
#pragma unroll
        for (int off = 16; off > 0; off >>= 1)
            wmax = fmaxf(wmax, __shfl_xor(wmax, off, 32));
        if (l == 0) smem[2 * ROW_LEN + p * 8 + w] = wmax;
        __syncthreads();
        float vmax = smem[2 * ROW_LEN + p * 8];
#pragma unroll
        for (int j = 1; j < 8; ++j)
            vmax = fmaxf(vmax, smem[2 * ROW_LEN + p * 8 + j]);

        // --- nano prescale: round row-max mantissa -> m_nano2 in {1,2} ---
        float m2 = 1.f;
        if (vmax > 0.f) {
            int   e = ilogb_pos(vmax);
            float m = ldexpf(vmax, -e);                 // [1,2)
            m  = floorf(m * 2.f + 0.5f) * 0.5f;         // f1 = 2
            m2 = floorf(m + 0.5f);                      // f2 = 1
        }
        const float inv_nano = (m2 > 1.5f) ? 0.5f : 1.f; // exact 1/m_nano2

        // --- block shared exponent ---
        float ba = bmax * inv_nano;
        int   se = (ba > 0.f) ? ilogb_pos(ba) : 0;
        se -= 2;                                        // EMAX = 2
        if (se < -127) se = -127;
        const float inv_s  = ldexpf(inv_nano, -se);     // inv_nano * 2^-se
        const float outmul = ldexpf(m2, se);            // m_nano2 * 2^se

        // --- quantize-dequantize 32 elements ---
#pragma unroll
        for (int k = 0; k < 8; ++k) {
            float4 q = v[k];
            q.x = qelem(q.x, inv_s, outmul);
            q.y = qelem(q.y, inv_s, outmul);
            q.z = qelem(q.z, inv_s, outmul);
            q.w = qelem(q.w, inv_s, outmul);
            v[k] = q;
        }

        // --- store ---
        float4* op = (float4*)(out + (size_t)r * ROW_LEN) + t * 8;
#pragma unroll
        for (int k = 0; k < 8; ++k) op[k] = v[k];
    }
}

extern "C" void kernel_launch(void* const* d_in, const int* in_sizes, int n_in,
                              void* d_out, int out_size, void* d_ws, size_t ws_size,
                              hipStream_t stream) {
    (void)n_in; (void)out_size; (void)d_ws; (void)ws_size;
    const float* x   = (const float*)d_in[0];
    float*       out = (float*)d_out;
    const long long n = (long long)in_sizes[0];
    const int nrows   = (int)(n / ROW_LEN);   // 8192 for the reference shape
    int grid = nrows < NUM_WG ? nrows : NUM_WG;
    if (grid < 1) grid = 1;
    mxq_kernel<<<grid, THREADS, 0, stream>>>(x, out, nrows);
}